// Encoding_14053132992759
// MI455X (gfx1250) — compile-verified
//
#include <hip/hip_runtime.h>
#include <hip/hip_bf16.h>

typedef __attribute__((ext_vector_type(2))) float v2f;
typedef __attribute__((ext_vector_type(8))) float v8f;
typedef __attribute__((ext_vector_type(4))) unsigned int v4u;
typedef __attribute__((ext_vector_type(8))) int v8i;
typedef __attribute__((ext_vector_type(4))) int v4i;

#define D_DIM 512
#define K_DIM 32
#define N_TOT 16384
#define NT 128
#define XS_STRIDE 136   // 128 + 8 : matches TDM pad_interval=6 (128 dw), pad_amount=7 (8 dw)
#define AS_STRIDE 33
#define BPB 32   // blocks per batch
#define IT 4     // 128-wide n-tiles per block  (BPB*IT*NT == N_TOT)

// workspace layout (floats)
#define WS_ASUM 0        // 8*32
#define WS_C2   256      // 32
#define WS_CT   512      // 512*32 transposed codewords Ct[d][k]

#if __has_builtin(__builtin_amdgcn_tensor_load_to_lds)
#define USE_TDM 1
#else
#define USE_TDM 0
#endif

// ---------------------------------------------------------------------------
// prep: zero output + asum, transpose codewords, compute c2
// ---------------------------------------------------------------------------
__global__ __launch_bounds__(256) void enc_prep(const float* __restrict__ C,
                                                float* __restrict__ out,
                                                float* __restrict__ ws) {
  const int tid = blockIdx.x * 256 + threadIdx.x;
  if (tid < 8 * K_DIM * D_DIM) out[tid] = 0.0f;
  if (tid < K_DIM * D_DIM) {
    const int d = tid >> 5;
    const int k = tid & 31;
    ws[WS_CT + tid] = C[k * D_DIM + d];          // Ct[d*32+k] = C[k][d]
  }
  if (tid < 8 * K_DIM) ws[WS_ASUM + tid] = 0.0f;
  if (tid < K_DIM) {
    float s = 0.0f;
    for (int d = 0; d < D_DIM; ++d) { const float c = C[tid * D_DIM + d]; s += c * c; }
    ws[WS_C2 + tid] = s;
  }
}

// ---------------------------------------------------------------------------
// main: single pass over X; GEMM1 + softmax + GEMM2 with X resident in LDS.
// X tile staged by the Tensor Data Mover (hardware-padded rows).
// ---------------------------------------------------------------------------
__global__ __launch_bounds__(256) void enc_main(const float* __restrict__ X,
                                                const float* __restrict__ scale,
                                                float* __restrict__ out,
                                                float* __restrict__ ws) {
  __shared__ float Xs[D_DIM * XS_STRIDE];   // 278528 B : Xs[d][n], padded rows
  __shared__ float As[NT * AS_STRIDE];      //  16896 B : A[n][k], padded
  __shared__ float x2p[256];
  __shared__ float x2s[NT];
  __shared__ float asum_s[K_DIM];

  const int t   = threadIdx.x;
  const int w   = t >> 5;            // wave 0..7
  const int l   = t & 31;
  const int l15 = l & 15;
  const int hi  = (l >> 4) & 1;      // upper half-wave?
  const int koff = hi ? 2 : 0;       // K-dim offset for A-cols / B-rows
  const int roff = hi ? 8 : 0;       // row offset for C/D fragments

  const int b     = blockIdx.x / BPB;
  const int chunk = blockIdx.x % BPB;

  const float* Xb  = X + (size_t)b * D_DIM * N_TOT;
  const float* Ctg = ws + WS_CT;

  const float sk0 = scale[l15];
  const float sk1 = scale[l15 + 16];
  const float c20 = ws[WS_C2 + l15];
  const float c21 = ws[WS_C2 + l15 + 16];

  if (t < K_DIM) asum_s[t] = 0.0f;

  v8f eacc[2][4];
#pragma unroll
  for (int kt = 0; kt < 2; ++kt)
#pragma unroll
    for (int j = 0; j < 4; ++j)
#pragma unroll
      for (int i = 0; i < 8; ++i) eacc[kt][j][i] = 0.0f;

  for (int it = 0; it < IT; ++it) {
    const int n0g = (chunk * IT + it) * NT;
    __syncthreads();                               // protect Xs/As reuse

#if USE_TDM
    // ---- stage X tile via Tensor Data Mover: 2-D tile (128 x 512 rows),
    //      hardware inserts 8 dwords of LDS padding every 128 dwords
    if (w == 0) {
      const unsigned long long ga = (unsigned long long)(uintptr_t)(Xb + n0g);
      v4u g0;
      g0[0] = 1u;                                   // count=1, no gather
      g0[1] = (unsigned int)(uintptr_t)&Xs[0];      // lds_addr (byte offset)
      g0[2] = (unsigned int)(ga & 0xffffffffu);     // global_addr[31:0]
      g0[3] = (unsigned int)((ga >> 32) & 0x01ffffffu) | (2u << 30); // addr[56:32], type=2
      v8i g1;
      g1[0] = (int)((2u << 16)    // data_size = 4B
                  | (1u << 20)    // pad_enable
                  | (6u << 22)    // pad_interval: 128 dwords
                  | (7u << 25));  // pad_amount  : 8 dwords
      g1[1] = (int)((unsigned)N_TOT << 16);         // tensor_dim0[15:0] in [31:16]
      g1[2] = (int)((unsigned)D_DIM << 16);         // tensor_dim0 hi=0 | tensor_dim1 lo
      g1[3] = (int)((unsigned)NT << 16);            // tensor_dim1 hi=0 | tile_dim0=128
      g1[4] = (int)D_DIM;                           // tile_dim1=512 | tile_dim2=0
      g1[5] = (int)N_TOT;                           // tensor_dim0_stride lo32
      g1[6] = 0;                                    // stride hi | dim1_stride lo
      g1[7] = 0;
      const v4i gz4 = {0, 0, 0, 0};
      const v8i gz8 = {0, 0, 0, 0, 0, 0, 0, 0};
      __builtin_amdgcn_tensor_load_to_lds(g0, g1, gz4, gz4, gz8, 0);
      __builtin_amdgcn_s_wait_tensorcnt(0);
    }
#else
    // ---- fallback: staged float4 copy through VGPRs
    for (int i = t; i < D_DIM * (NT / 4); i += 256) {
      const int d  = i >> 5;
      const int c4 = (i & 31) << 2;
      const float4 v = *(const float4*)(Xb + (size_t)d * N_TOT + n0g + c4);
      *(float4*)(&Xs[d * XS_STRIDE + c4]) = v;
    }
#endif
    __syncthreads();

    // ---- x2[n] = sum_d Xs[d][n]^2  (two half-D partials per n)
    {
      const int n = t & (NT - 1);
      const int h = t >> 7;
      float s = 0.0f;
      for (int d = h * 256; d < h * 256 + 256; ++d) {
        const float x = Xs[d * XS_STRIDE + n];
        s += x * x;
      }
      x2p[t] = s;
    }
    __syncthreads();
    if (t < NT) x2s[t] = x2p[t] + x2p[t + NT];

    // ---- GEMM1: xc[n,k] tile per wave (n-subtile w, k-tiles 0 and 1)
    v8f acc0, acc1;
#pragma unroll
    for (int i = 0; i < 8; ++i) { acc0[i] = 0.0f; acc1[i] = 0.0f; }
    {
      const int arow = w * 16 + l15;               // n row of A fragment
      for (int d0 = 0; d0 < D_DIM; d0 += 4) {
        v2f a, b0, b1;
        a.x  = Xs[(d0 + koff) * XS_STRIDE + arow];
        a.y  = Xs[(d0 + koff + 1) * XS_STRIDE + arow];
        b0.x = Ctg[(d0 + koff) * K_DIM + l15];
        b0.y = Ctg[(d0 + koff + 1) * K_DIM + l15];
        b1.x = Ctg[(d0 + koff) * K_DIM + 16 + l15];
        b1.y = Ctg[(d0 + koff + 1) * K_DIM + 16 + l15];
        acc0 = __builtin_amdgcn_wmma_f32_16x16x4_f32(false, a, false, b0,
                                                     (short)0, acc0, false, false);
        acc1 = __builtin_amdgcn_wmma_f32_16x16x4_f32(false, a, false, b1,
                                                     (short)0, acc1, false, false);
      }
    }
    __syncthreads();                               // x2s ready

    // ---- softmax over k (32 values live across 16 lanes x 2 accumulators)
    {
      const int nbase = w * 16 + roff;
      float pa0 = 0.0f, pa1 = 0.0f;
#pragma unroll
      for (int i = 0; i < 8; ++i) {
        const int n = nbase + i;
        const float x2 = x2s[n];
        float sl0 = sk0 * (x2 - 2.0f * acc0[i] + c20);
        float sl1 = sk1 * (x2 - 2.0f * acc1[i] + c21);
        float m = fmaxf(sl0, sl1);
#pragma unroll
        for (int off = 8; off >= 1; off >>= 1)
          m = fmaxf(m, __shfl_xor(m, off, 32));
        const float e0 = __expf(sl0 - m);
        const float e1 = __expf(sl1 - m);
        float s = e0 + e1;
#pragma unroll
        for (int off = 8; off >= 1; off >>= 1)
          s += __shfl_xor(s, off, 32);
        const float r = __builtin_amdgcn_rcpf(s);
        const float a0 = e0 * r;
        const float a1 = e1 * r;
        As[n * AS_STRIDE + l15]      = a0;
        As[n * AS_STRIDE + 16 + l15] = a1;
        pa0 += a0; pa1 += a1;
      }
      pa0 += __shfl_xor(pa0, 16, 32);
      pa1 += __shfl_xor(pa1, 16, 32);
      if (!hi) {
        atomicAdd(&asum_s[l15], pa0);
        atomicAdd(&asum_s[l15 + 16], pa1);
      }
    }
    __syncthreads();                               // As complete

    // ---- GEMM2: E1[k,d] += A^T * Xf  (accumulators persist across iters)
    {
      for (int nl = 0; nl < NT; nl += 4) {
        v2f a0, a1;
        a0.x = As[(nl + koff) * AS_STRIDE + l15];
        a0.y = As[(nl + koff + 1) * AS_STRIDE + l15];
        a1.x = As[(nl + koff) * AS_STRIDE + 16 + l15];
        a1.y = As[(nl + koff + 1) * AS_STRIDE + 16 + l15];
#pragma unroll
        for (int j = 0; j < 4; ++j) {
          const int dt = w * 4 + j;
          const v2f bb = *(const v2f*)(&Xs[(dt * 16 + l15) * XS_STRIDE + nl + koff]);
          eacc[0][j] = __builtin_amdgcn_wmma_f32_16x16x4_f32(false, a0, false, bb,
                                                             (short)0, eacc[0][j], false, false);
          eacc[1][j] = __builtin_amdgcn_wmma_f32_16x16x4_f32(false, a1, false, bb,
                                                             (short)0, eacc[1][j], false, false);
        }
      }
    }
  }

  __syncthreads();
  // ---- commit partials
  {
    float* ob = out + (size_t)b * K_DIM * D_DIM;
#pragma unroll
    for (int kt = 0; kt < 2; ++kt)
#pragma unroll
      for (int j = 0; j < 4; ++j) {
        const int dcol = (w * 4 + j) * 16 + l15;
#pragma unroll
        for (int i = 0; i < 8; ++i) {
          const int krow = kt * 16 + i + roff;
          atomicAdd(&ob[krow * D_DIM + dcol], eacc[kt][j][i]);
        }
      }
    if (t < K_DIM) atomicAdd(&ws[WS_ASUM + b * K_DIM + t], asum_s[t]);
  }
}

// ---------------------------------------------------------------------------
// finalize: E[b,k,d] -= asum[b,k] * C[k,d]
// ---------------------------------------------------------------------------
__global__ __launch_bounds__(128) void enc_fin(const float* __restrict__ C,
                                               float* __restrict__ out,
                                               const float* __restrict__ ws) {
  const int bk = blockIdx.x;           // 0..255
  const int k  = bk & 31;
  const float a = ws[WS_ASUM + bk];
  float* o = out + (size_t)bk * D_DIM;
  const float* c = C + (size_t)k * D_DIM;
  for (int d = threadIdx.x; d < D_DIM; d += 128) o[d] -= a * c[d];
}

extern "C" void kernel_launch(void* const* d_in, const int* in_sizes, int n_in,
                              void* d_out, int out_size, void* d_ws, size_t ws_size,
                              hipStream_t stream) {
  const float* X  = (const float*)d_in[0];   // [8,512,128,128]
  const float* C  = (const float*)d_in[1];   // [32,512]
  const float* S  = (const float*)d_in[2];   // [32]
  float* out = (float*)d_out;                // [8,32,512]
  float* ws  = (float*)d_ws;

  enc_prep<<<512, 256, 0, stream>>>(C, out, ws);
  enc_main<<<8 * BPB, 256, 0, stream>>>(X, S, out, ws);
  enc_fin<<<8 * K_DIM, 128, 0, stream>>>(C, out, ws);
}